// SupplyChainGNN_14980845929055
// MI455X (gfx1250) — compile-verified
//
#include <hip/hip_runtime.h>
#include <hip/hip_bf16.h>

typedef __attribute__((ext_vector_type(2))) float v2f;
typedef __attribute__((ext_vector_type(8))) float v8f;

// ---------------------------------------------------------------------------
// Degree / normalization precompute
// ---------------------------------------------------------------------------
__global__ __launch_bounds__(256) void deg_init_kernel(float* __restrict__ deg, int n) {
    int i = blockIdx.x * 256 + threadIdx.x;
    if (i < n) deg[i] = 1.0f;  // self-loop contributes 1
}

__global__ __launch_bounds__(256) void deg_count_kernel(float* __restrict__ deg,
                                                        const long long* __restrict__ dst,
                                                        long long nE) {
    long long e = (long long)blockIdx.x * 256 + threadIdx.x;
    if (e < nE) atomicAdd(&deg[(int)dst[e]], 1.0f);
}

__global__ __launch_bounds__(256) void deg_rsqrt_kernel(float* __restrict__ deg, int n) {
    int i = blockIdx.x * 256 + threadIdx.x;
    if (i < n) deg[i] = rsqrtf(fmaxf(deg[i], 1.0f));  // in-place: deg -> dinv
}

// Pack edges to int32 + precompute per-edge symmetric norm (reused by 3 layers)
__global__ __launch_bounds__(256) void edge_prep_kernel(const long long* __restrict__ src,
                                                        const long long* __restrict__ dst,
                                                        const float* __restrict__ dinv,
                                                        int* __restrict__ esrc,
                                                        int* __restrict__ edst,
                                                        float* __restrict__ enorm,
                                                        long long nE) {
    long long e = (long long)blockIdx.x * 256 + threadIdx.x;
    if (e >= nE) return;
    int s = (int)src[e];
    int d = (int)dst[e];
    esrc[e] = s;
    edst[e] = d;
    enorm[e] = dinv[s] * dinv[d];
}

// ---------------------------------------------------------------------------
// GEMM: G[nRows x NOUT] = act(X[nRows x 128]) @ W[128 x NOUT]   (fp32 WMMA)
// Block = 256 threads = 8 waves; each wave owns a 16-row strip x all NOUT cols.
// Weights staged in LDS in exact B-fragment order:
//   Wf[((t*32 + k0/4)*32 + lane)*2 + j] = W[k][n],
//   n = t*16 + (lane&15), k = k0 + 2*(lane>>4) + j
// so a fragment load is one conflict-free ds_load_b64 (64 banks hit once).
// ---------------------------------------------------------------------------
template <int NOUT, bool RELU>
__global__ __launch_bounds__(256) void gemm_wmma_kernel(const float* __restrict__ X,
                                                        const float* __restrict__ W,
                                                        float* __restrict__ G, int nRows) {
    constexpr int K  = 128;
    constexpr int NT = NOUT / 16;
    __shared__ float Wf[NOUT * K];  // 64KB (NOUT=128) / 16KB (NOUT=32)

    const int tid = threadIdx.x;
    for (int s = tid; s < NOUT * K; s += 256) {
        int j     = s & 1;
        int l     = (s >> 1) & 31;
        int kstep = (s >> 6) & 31;
        int t     = s >> 11;
        int n = t * 16 + (l & 15);
        int k = kstep * 4 + ((l >> 4) << 1) + j;
        Wf[s] = W[k * NOUT + n];
    }
    __syncthreads();

    const int lane    = tid & 31;
    const int wave    = tid >> 5;
    const int rowBase = blockIdx.x * 128 + wave * 16;
    int m = rowBase + (lane & 15);
    if (m > nRows - 1) m = nRows - 1;  // clamp; duplicate rows masked at store
    const int khalf = (lane >> 4) << 1;  // 0 or 2
    const float* __restrict__ xrow = X + (size_t)m * K;

    const v8f vzero = {0.f, 0.f, 0.f, 0.f, 0.f, 0.f, 0.f, 0.f};
    v8f acc[NT];
#pragma unroll
    for (int t = 0; t < NT; ++t) acc[t] = vzero;

#pragma unroll 4
    for (int k0 = 0; k0 < K; k0 += 4) {
        // A fragment: lane(0..15)->rows, khalf selects K pair -> aligned b64 load
        v2f a = *(const v2f*)(xrow + k0 + khalf);
        if (RELU) {
            a.x = fmaxf(a.x, 0.f);
            a.y = fmaxf(a.y, 0.f);
        }
#pragma unroll
        for (int t = 0; t < NT; ++t) {
            v2f b = *(const v2f*)(Wf + (((t * 32 + (k0 >> 2)) * 32 + lane) << 1));
            acc[t] = __builtin_amdgcn_wmma_f32_16x16x4_f32(false, a, false, b, (short)0,
                                                           acc[t], false, false);
        }
    }

    // C/D layout: VGPR v holds row rowBase + v + 8*(lane>=16), col = tile*16 + (lane&15)
    const int rhalf = (lane >> 4) << 3;
#pragma unroll
    for (int t = 0; t < NT; ++t) {
        const int col = t * 16 + (lane & 15);
#pragma unroll
        for (int v = 0; v < 8; ++v) {
            int row = rowBase + rhalf + v;
            if (row < nRows) G[(size_t)row * NOUT + col] = acc[t][v];
        }
    }
}

// ---------------------------------------------------------------------------
// Aggregation init: out[i][f] = bias[f] + g[i][f] * dinv[i]^2   (self-loop msg)
// ---------------------------------------------------------------------------
template <int F>
__global__ __launch_bounds__(256) void agg_init_kernel(const float* __restrict__ g,
                                                       const float* __restrict__ dinv,
                                                       const float* __restrict__ bias,
                                                       float* __restrict__ out, int n) {
    size_t i = (size_t)blockIdx.x * 256 + threadIdx.x;
    if (i >= (size_t)n * F) return;
    int node = (int)(i / F);
    int f    = (int)(i % F);
    float d  = dinv[node];
    out[i] = bias[f] + g[i] * d * d;
}

// ---------------------------------------------------------------------------
// Edge scatter: warp per edge. float4 gather of h[src] (L2-hot), scale by norm,
// float atomic-add into out[dst].
// ---------------------------------------------------------------------------
template <int F>
__global__ __launch_bounds__(256) void agg_edge_kernel(const float* __restrict__ g,
                                                       const int* __restrict__ esrc,
                                                       const int* __restrict__ edst,
                                                       const float* __restrict__ enorm,
                                                       float* __restrict__ out, long long nE) {
    long long e = (long long)blockIdx.x * 8 + (threadIdx.x >> 5);
    if (e >= nE) return;
    const int lane = threadIdx.x & 31;
    const int s = esrc[e];
    const int d = edst[e];
    const float w = enorm[e];
    if constexpr (F == 128) {
        float4 v = *((const float4*)(g + (size_t)s * 128) + lane);
        float* o = out + (size_t)d * 128 + lane * 4;
        atomicAdd(o + 0, v.x * w);
        atomicAdd(o + 1, v.y * w);
        atomicAdd(o + 2, v.z * w);
        atomicAdd(o + 3, v.w * w);
    } else {
        atomicAdd(out + (size_t)d * F + lane, g[(size_t)s * F + lane] * w);
    }
}

// ---------------------------------------------------------------------------
// Global mean pool over batch ids (64 graphs x 32 feats)
// ---------------------------------------------------------------------------
__global__ __launch_bounds__(256) void pool_zero_kernel(float* __restrict__ sums,
                                                        float* __restrict__ cnts) {
    int i = blockIdx.x * 256 + threadIdx.x;
    if (i < 64 * 32) sums[i] = 0.f;
    if (i < 64) cnts[i] = 0.f;
}

__global__ __launch_bounds__(256) void pool_accum_kernel(const float* __restrict__ h,
                                                         const long long* __restrict__ batch,
                                                         float* __restrict__ sums,
                                                         float* __restrict__ cnts, int n) {
    int node = blockIdx.x * 8 + (threadIdx.x >> 5);
    if (node >= n) return;
    int lane = threadIdx.x & 31;
    int gid  = (int)batch[node];
    atomicAdd(&sums[(gid << 5) + lane], h[(size_t)node * 32 + lane]);
    if (lane == 0) atomicAdd(&cnts[gid], 1.0f);
}

__global__ __launch_bounds__(256) void pool_final_kernel(const float* __restrict__ sums,
                                                         const float* __restrict__ cnts,
                                                         float* __restrict__ xg) {
    int i = blockIdx.x * 256 + threadIdx.x;
    if (i < 64 * 32) xg[i] = sums[i] / fmaxf(cnts[i >> 5], 1.0f);
}

// ---------------------------------------------------------------------------
// Host launcher
// ---------------------------------------------------------------------------
extern "C" void kernel_launch(void* const* d_in, const int* in_sizes, int n_in,
                              void* d_out, int out_size, void* d_ws, size_t ws_size,
                              hipStream_t stream) {
    const float* x        = (const float*)d_in[0];
    const long long* eidx = (const long long*)d_in[1];
    const long long* batch = (const long long*)d_in[2];
    const float* W1 = (const float*)d_in[4];
    const float* b1 = (const float*)d_in[5];
    const float* W2 = (const float*)d_in[6];
    const float* b2 = (const float*)d_in[7];
    const float* W3 = (const float*)d_in[8];
    const float* b3 = (const float*)d_in[9];

    const int       N = in_sizes[0] / 128;
    const long long E = in_sizes[1] / 2;
    const long long* srcL = eidx;
    const long long* dstL = eidx + E;

    // Carve workspace (16-float / 64B aligned chunks)
    float* wsf = (float*)d_ws;
    size_t off = 0;
    auto carve = [&](size_t nElems) -> float* {
        float* p = wsf + off;
        off += (nElems + 15) & ~(size_t)15;
        return p;
    };
    float* dinv  = carve((size_t)N);            // deg -> dinv in place
    float* g     = carve((size_t)N * 128);      // GEMM output
    float* a     = carve((size_t)N * 128);      // aggregated output (layer input)
    float* enorm = carve((size_t)E);
    int*   esrc  = (int*)carve((size_t)E);
    int*   edst  = (int*)carve((size_t)E);
    float* sums  = carve(64 * 32);
    float* cnts  = carve(64);

    float* hout = (float*)d_out;                // [N x 32]
    float* xg   = hout + (size_t)N * 32;        // [64 x 32]

    const int nb256_N   = (N + 255) / 256;
    const int nb256_E   = (int)((E + 255) / 256);
    const int nbGemm    = (N + 127) / 128;
    const int nbEdge    = (int)((E + 7) / 8);
    const int nbF128    = (int)(((size_t)N * 128 + 255) / 256);
    const int nbF32     = (int)(((size_t)N * 32 + 255) / 256);
    const int nbPoolAcc = (N + 7) / 8;

    // --- normalization precompute ---
    deg_init_kernel<<<nb256_N, 256, 0, stream>>>(dinv, N);
    deg_count_kernel<<<nb256_E, 256, 0, stream>>>(dinv, dstL, E);
    deg_rsqrt_kernel<<<nb256_N, 256, 0, stream>>>(dinv, N);
    edge_prep_kernel<<<nb256_E, 256, 0, stream>>>(srcL, dstL, dinv, esrc, edst, enorm, E);

    // --- layer 1: x(128) -> a(128) ---
    gemm_wmma_kernel<128, false><<<nbGemm, 256, 0, stream>>>(x, W1, g, N);
    agg_init_kernel<128><<<nbF128, 256, 0, stream>>>(g, dinv, b1, a, N);
    agg_edge_kernel<128><<<nbEdge, 256, 0, stream>>>(g, esrc, edst, enorm, a, E);

    // --- layer 2: relu(a)(128) -> a(128)  (relu fused into GEMM A-load) ---
    gemm_wmma_kernel<128, true><<<nbGemm, 256, 0, stream>>>(a, W2, g, N);
    agg_init_kernel<128><<<nbF128, 256, 0, stream>>>(g, dinv, b2, a, N);
    agg_edge_kernel<128><<<nbEdge, 256, 0, stream>>>(g, esrc, edst, enorm, a, E);

    // --- layer 3: relu(a)(128) -> hout(32) ---
    gemm_wmma_kernel<32, true><<<nbGemm, 256, 0, stream>>>(a, W3, g, N);
    agg_init_kernel<32><<<nbF32, 256, 0, stream>>>(g, dinv, b3, hout, N);
    agg_edge_kernel<32><<<nbEdge, 256, 0, stream>>>(g, esrc, edst, enorm, hout, E);

    // --- global mean pool ---
    pool_zero_kernel<<<8, 256, 0, stream>>>(sums, cnts);
    pool_accum_kernel<<<nbPoolAcc, 256, 0, stream>>>(hout, batch, sums, cnts, N);
    pool_final_kernel<<<8, 256, 0, stream>>>(sums, cnts, xg);
}